// SlotAttention_81784767251084
// MI455X (gfx1250) — compile-verified
//
#include <hip/hip_runtime.h>
#include <math.h>

typedef float v2f __attribute__((ext_vector_type(2)));
typedef float v8f __attribute__((ext_vector_type(8)));

#define EPS_ATTN 1e-6f
#define LN_EPS   1e-5f

__device__ __forceinline__ v8f wmma_f32_16x16x4(v2f a, v2f b, v8f c) {
  // D = A(16x4 f32) * B(4x16 f32) + C(16x16 f32)
  return __builtin_amdgcn_wmma_f32_16x16x4_f32(false, a, false, b, (short)0, c,
                                               false, false);
}

// Issue one per-lane async 16B copy global -> LDS (tracked by ASYNCcnt).
// LDS byte offset = low 32 bits of the generic address.
__device__ __forceinline__ void async_copy_b128(const float* gp, void* lds) {
  unsigned int loff = (unsigned int)(size_t)lds;
  asm volatile("global_load_async_to_lds_b128 %0, %1, off"
               :
               : "v"(loff), "v"(gp)
               : "memory");
}
__device__ __forceinline__ void wait_asynccnt0() {
  asm volatile("s_wait_asynccnt 0x0" ::: "memory");
}

// ---------------------------------------------------------------------------
// Kernel A: fused LayerNorm(inputs) + k = x@Wk^T, v = x@Wv^T
// Grid: 2048 WGs x 256 thr. Each WG: 64 rows of (B*N, 512) -> k,v (64 x 128).
// K chunked by 32; weight chunk (both matrices) async-staged in LDS.
// ---------------------------------------------------------------------------
__global__ __launch_bounds__(256) void k_ln_kv(
    const float* __restrict__ inp, const float* __restrict__ lw,
    const float* __restrict__ lb, const float* __restrict__ Wk,
    const float* __restrict__ Wv, float* __restrict__ kbuf,
    float* __restrict__ vbuf) {
  constexpr int D = 512;
  __shared__ float xt[64][36];    // normalized x chunk, padded stride
  __shared__ float wt[256][36];   // concat(Wk,Wv) chunk, [out][k], padded
  __shared__ float sred[64][8];
  __shared__ float smu[64], srs[64];

  const int tid = threadIdx.x;
  const size_t rowBase = (size_t)blockIdx.x * 64;

  // ---- phase 1: per-row mean/var over D=512 (4 threads per row) ----
  {
    int row = tid >> 2, part = tid & 3;
    const float* xr = inp + (rowBase + row) * D + part * 4;
    float s = 0.f, ss = 0.f;
#pragma unroll
    for (int i = 0; i < 32; ++i) {
      float4 t = *(const float4*)(xr + i * 16);
      s += t.x + t.y + t.z + t.w;
      ss += t.x * t.x + t.y * t.y + t.z * t.z + t.w * t.w;
    }
    sred[row][part] = s;
    sred[row][part + 4] = ss;
  }
  __syncthreads();
  if (tid < 64) {
    float s = sred[tid][0] + sred[tid][1] + sred[tid][2] + sred[tid][3];
    float ss = sred[tid][4] + sred[tid][5] + sred[tid][6] + sred[tid][7];
    float mu = s * (1.f / 512.f);
    float var = ss * (1.f / 512.f) - mu * mu;
    smu[tid] = mu;
    srs[tid] = rsqrtf(var + LN_EPS);
  }
  __syncthreads();

  const int l = tid & 31, wv = tid >> 5;
  const int strip = wv & 3;      // which 16-row strip
  const int halfsel = wv >> 2;   // 0 -> k, 1 -> v
  float* dst = halfsel ? vbuf : kbuf;
  const int mrow = strip * 16 + (l & 15);
  const int kko = (l >> 4) * 2;  // lane-half K offset
  const int nl = l & 15;
  const int wrb = halfsel * 128; // weight-row base inside wt

  v8f zero = {};
  v8f acc[8];
#pragma unroll
  for (int ct = 0; ct < 8; ++ct) acc[ct] = zero;

  // ---- phase 2: K chunks of 32; weights via async-to-LDS, x via VALU LN ----
  for (int kc = 0; kc < 16; ++kc) {
    const int kbase = kc * 32;
    // async-stage weight chunk: 256 outs x 32 k = 2048 float4 issues
#pragma unroll
    for (int j = 0; j < 8; ++j) {
      int idx = tid + j * 256;
      int row = idx >> 3, cg = idx & 7;
      const float* gp = (row < 128)
                            ? (Wk + (size_t)row * D + kbase + cg * 4)
                            : (Wv + (size_t)(row - 128) * D + kbase + cg * 4);
      async_copy_b128(gp, &wt[row][cg * 4]);
    }
    // overlap: normalize x chunk into LDS while async copies fly
#pragma unroll
    for (int j = 0; j < 2; ++j) {
      int idx = tid + j * 256;
      int row = idx >> 3, cg = idx & 7;
      int c0 = cg * 4, gc = kbase + c0;
      float4 t = *(const float4*)(inp + (rowBase + row) * D + gc);
      float mu = smu[row], rs = srs[row];
      xt[row][c0 + 0] = (t.x - mu) * rs * lw[gc + 0] + lb[gc + 0];
      xt[row][c0 + 1] = (t.y - mu) * rs * lw[gc + 1] + lb[gc + 1];
      xt[row][c0 + 2] = (t.z - mu) * rs * lw[gc + 2] + lb[gc + 2];
      xt[row][c0 + 3] = (t.w - mu) * rs * lw[gc + 3] + lb[gc + 3];
    }
    wait_asynccnt0();
    __syncthreads();
#pragma unroll
    for (int k0 = 0; k0 < 32; k0 += 4) {
      v2f a;
      a.x = xt[mrow][k0 + kko];
      a.y = xt[mrow][k0 + kko + 1];
#pragma unroll
      for (int ct = 0; ct < 8; ++ct) {
        const float* wp = &wt[wrb + ct * 16 + nl][k0 + kko];
        v2f b;
        b.x = wp[0];
        b.y = wp[1];
        acc[ct] = wmma_f32_16x16x4(a, b, acc[ct]);
      }
    }
    __syncthreads();
  }

  // ---- store: C layout M = r + 8*(lane>=16), N = lane%16 ----
  const int mb = (l >> 4) * 8;
#pragma unroll
  for (int ct = 0; ct < 8; ++ct) {
#pragma unroll
    for (int r = 0; r < 8; ++r) {
      size_t g = rowBase + strip * 16 + mb + r;
      dst[g * 128 + ct * 16 + nl] = acc[ct][r];
    }
  }
}

// ---------------------------------------------------------------------------
// Kernel B: slots = mu + exp(log_sigma) * slots_init    (32*8*128 elems)
// ---------------------------------------------------------------------------
__global__ __launch_bounds__(256) void k_init_slots(
    const float* __restrict__ init, const float* __restrict__ mu,
    const float* __restrict__ ls, float* __restrict__ slots) {
  int idx = blockIdx.x * 256 + threadIdx.x;
  if (idx < 32 * 8 * 128) {
    int s = idx & 127;
    slots[idx] = mu[s] + __expf(ls[s]) * init[idx];
  }
}

// ---------------------------------------------------------------------------
// Kernel C: q = scale * LN(slots) @ Wq^T   (256 rows x 128)
// Grid: 16 WGs x 256 thr; 16 rows per WG; wave w computes cols [16w,16w+16)
// ---------------------------------------------------------------------------
__global__ __launch_bounds__(256) void k_q(
    const float* __restrict__ slots, const float* __restrict__ lw,
    const float* __restrict__ lb, const float* __restrict__ Wq,
    float* __restrict__ q) {
  constexpr int S = 128;
  __shared__ float sl[16][132];
  __shared__ float sred[16][32];
  __shared__ float smu[16], srs[16];
  int tid = threadIdx.x;
  int rowBase = blockIdx.x * 16;
  int row = tid >> 4, part = tid & 15;
  float vals[8];
  {
    const float* xr = slots + (size_t)(rowBase + row) * S + part * 8;
    float s = 0.f, ss = 0.f;
#pragma unroll
    for (int j = 0; j < 8; ++j) {
      float x = xr[j];
      vals[j] = x;
      s += x;
      ss += x * x;
    }
    sred[row][part] = s;
    sred[row][part + 16] = ss;
  }
  __syncthreads();
  if (tid < 16) {
    float s = 0.f, ss = 0.f;
    for (int p = 0; p < 16; ++p) {
      s += sred[tid][p];
      ss += sred[tid][p + 16];
    }
    float mu = s * (1.f / 128.f);
    float var = ss * (1.f / 128.f) - mu * mu;
    smu[tid] = mu;
    srs[tid] = rsqrtf(var + LN_EPS);
  }
  __syncthreads();
  {
    float mu = smu[row], rs = srs[row];
#pragma unroll
    for (int j = 0; j < 8; ++j) {
      int c = part * 8 + j;
      sl[row][c] = (vals[j] - mu) * rs * lw[c] + lb[c];
    }
  }
  __syncthreads();
  int l = tid & 31, wv = tid >> 5;
  int mr = l & 15, kko = (l >> 4) * 2, nl = l & 15;
  v8f acc = {};
  for (int k0 = 0; k0 < 128; k0 += 4) {
    v2f a;
    a.x = sl[mr][k0 + kko];
    a.y = sl[mr][k0 + kko + 1];
    const float* wp = Wq + (size_t)(wv * 16 + nl) * S + k0 + kko;
    v2f b;
    b.x = wp[0];
    b.y = wp[1];
    acc = wmma_f32_16x16x4(a, b, acc);
  }
  const float scale = 0.08838834764831845f;  // 128^-0.5 folded into q
  int mb = (l >> 4) * 8;
#pragma unroll
  for (int r = 0; r < 8; ++r)
    q[(size_t)(rowBase + mb + r) * S + wv * 16 + nl] = acc[r] * scale;
}

// ---------------------------------------------------------------------------
// Kernel D: logits = k . q^T, softmax over 8 slots, +eps, store attn (B,N,8)
// Grid: 512 WGs x 256 thr; one input row per thread
// ---------------------------------------------------------------------------
__global__ __launch_bounds__(256) void k_attn(
    const float* __restrict__ kbuf, const float* __restrict__ q,
    float* __restrict__ attn) {
  __shared__ float qs[8][128];
  int tid = threadIdx.x;
  int b = blockIdx.x >> 4;
  int n = (blockIdx.x & 15) * 256 + tid;
  {
    float4 t = *(const float4*)(q + (size_t)b * 8 * 128 + tid * 4);
    int m = tid >> 5, c = (tid & 31) * 4;
    qs[m][c] = t.x;
    qs[m][c + 1] = t.y;
    qs[m][c + 2] = t.z;
    qs[m][c + 3] = t.w;
  }
  __syncthreads();
  const float4* kr = (const float4*)(kbuf + ((size_t)b * 4096 + n) * 128);
  float dot[8];
#pragma unroll
  for (int m = 0; m < 8; ++m) dot[m] = 0.f;
  for (int i = 0; i < 32; ++i) {
    float4 kv = kr[i];
#pragma unroll
    for (int m = 0; m < 8; ++m) {
      const float* qp = &qs[m][i * 4];
      dot[m] += kv.x * qp[0] + kv.y * qp[1] + kv.z * qp[2] + kv.w * qp[3];
    }
  }
  float mx = dot[0];
#pragma unroll
  for (int m = 1; m < 8; ++m) mx = fmaxf(mx, dot[m]);
  float e[8], sum = 0.f;
#pragma unroll
  for (int m = 0; m < 8; ++m) {
    e[m] = __expf(dot[m] - mx);
    sum += e[m];
  }
  float inv = 1.f / sum;
  float* ap = attn + ((size_t)b * 4096 + n) * 8;
  float4 o0 = {e[0] * inv + EPS_ATTN, e[1] * inv + EPS_ATTN,
               e[2] * inv + EPS_ATTN, e[3] * inv + EPS_ATTN};
  float4 o1 = {e[4] * inv + EPS_ATTN, e[5] * inv + EPS_ATTN,
               e[6] * inv + EPS_ATTN, e[7] * inv + EPS_ATTN};
  ((float4*)ap)[0] = o0;
  ((float4*)ap)[1] = o1;
}

// ---------------------------------------------------------------------------
// Kernel E: U[b,m,s] = (1/colsum[b,m]) * sum_n attn[b,n,m] * v[b,n,s]
// Grid: 32 WGs (one per batch) x 256 thr. attn^T as WMMA-A (M=8 padded to 16),
// v as B, K=4096 chunked through LDS; colsum via deterministic block reduce.
// ---------------------------------------------------------------------------
__global__ __launch_bounds__(256) void k_updates(
    const float* __restrict__ attn, const float* __restrict__ vbuf,
    float* __restrict__ U) {
  __shared__ float at[1024 * 8];   // attn chunk: 1024 rows x 8 slots
  __shared__ float red[256][8];
  int tid = threadIdx.x;
  int b = blockIdx.x;
  int l = tid & 31, wv = tid >> 5;
  int mr = l & 15, kko = (l >> 4) * 2, nl = l & 15;
  int ct = wv;  // column tile: cols [16*wv, 16*wv+16)
  v8f acc = {};
  float cs[8];
#pragma unroll
  for (int m = 0; m < 8; ++m) cs[m] = 0.f;

  for (int chunk = 0; chunk < 4; ++chunk) {
    int n0 = chunk * 1024;
    __syncthreads();
#pragma unroll
    for (int j = 0; j < 8; ++j) {
      int idx4 = tid + j * 256;
      float4 t =
          *(const float4*)(attn + ((size_t)b * 4096 + n0) * 8 + idx4 * 4);
      ((float4*)at)[idx4] = t;
    }
    __syncthreads();
#pragma unroll
    for (int rr = 0; rr < 4; ++rr) {
      const float* ar = at + (tid + rr * 256) * 8;
#pragma unroll
      for (int m = 0; m < 8; ++m) cs[m] += ar[m];
    }
    for (int k0 = 0; k0 < 1024; k0 += 4) {
      int kl = k0 + kko;
      v2f a;
      a.x = (mr < 8) ? at[kl * 8 + mr] : 0.f;        // A[m][k] = attn[n][m]
      a.y = (mr < 8) ? at[(kl + 1) * 8 + mr] : 0.f;
      const float* vp =
          vbuf + ((size_t)b * 4096 + n0 + kl) * 128 + ct * 16 + nl;
      v2f bb;
      bb.x = vp[0];
      bb.y = vp[128];  // next K row
      acc = wmma_f32_16x16x4(a, bb, acc);
    }
  }

  // deterministic colsum reduction
#pragma unroll
  for (int m = 0; m < 8; ++m) red[tid][m] = cs[m];
  __syncthreads();
  for (int s = 128; s > 0; s >>= 1) {
    if (tid < s) {
#pragma unroll
      for (int m = 0; m < 8; ++m) red[tid][m] += red[tid + s][m];
    }
    __syncthreads();
  }
  // lanes 0-15 hold rows M=0..7 (valid); lanes 16-31 hold padded rows 8..15
  if ((l >> 4) == 0) {
#pragma unroll
    for (int r = 0; r < 8; ++r)
      U[((size_t)b * 8 + r) * 128 + ct * 16 + nl] = acc[r] / red[0][r];
  }
}

// ---------------------------------------------------------------------------
// Kernel F: GRU cell + LN + 2-layer MLP + residual, one (b,m) row per WG
// Grid: 256 WGs x 128 thr
// ---------------------------------------------------------------------------
__global__ __launch_bounds__(128) void k_gru_mlp(
    const float* __restrict__ U, const float* __restrict__ slots,
    const float* __restrict__ Wih, const float* __restrict__ Whh,
    const float* __restrict__ bih, const float* __restrict__ bhh,
    const float* __restrict__ lnw, const float* __restrict__ lnb,
    const float* __restrict__ W1, const float* __restrict__ b1,
    const float* __restrict__ W2, const float* __restrict__ b2,
    float* __restrict__ dst) {
  __shared__ float su[128], sh[128], sy[128], shid[256], sA[128], sB[128];
  int t = threadIdx.x;
  size_t rb = (size_t)blockIdx.x * 128;
  float u = U[rb + t], hp = slots[rb + t];
  su[t] = u;
  sh[t] = hp;
  __syncthreads();
  float gi[3], gh[3];
#pragma unroll
  for (int g = 0; g < 3; ++g) {
    int j = g * 128 + t;
    float di = bih[j], dh = bhh[j];
    const float* wi = Wih + (size_t)j * 128;
    const float* wh = Whh + (size_t)j * 128;
    for (int i = 0; i < 128; ++i) {
      di += su[i] * wi[i];
      dh += sh[i] * wh[i];
    }
    gi[g] = di;
    gh[g] = dh;
  }
  float r = 1.f / (1.f + __expf(-(gi[0] + gh[0])));
  float z = 1.f / (1.f + __expf(-(gi[1] + gh[1])));
  float nn = tanhf(gi[2] + r * gh[2]);
  float h = (1.f - z) * nn + z * hp;
  sA[t] = h;
  sB[t] = h * h;
  __syncthreads();
  for (int s = 64; s > 0; s >>= 1) {
    if (t < s) {
      sA[t] += sA[t + s];
      sB[t] += sB[t + s];
    }
    __syncthreads();
  }
  float mu = sA[0] * (1.f / 128.f);
  float var = sB[0] * (1.f / 128.f) - mu * mu;
  float rs = rsqrtf(var + LN_EPS);
  sy[t] = (h - mu) * rs * lnw[t] + lnb[t];
  __syncthreads();
#pragma unroll
  for (int g = 0; g < 2; ++g) {
    int j = g * 128 + t;
    float d = b1[j];
    const float* w = W1 + (size_t)j * 128;
    for (int i = 0; i < 128; ++i) d += sy[i] * w[i];
    shid[j] = fmaxf(d, 0.f);
  }
  __syncthreads();
  float o = b2[t];
  const float* w2 = W2 + (size_t)t * 256;
  for (int j = 0; j < 256; ++j) o += shid[j] * w2[j];
  dst[rb + t] = h + o;
}

// ---------------------------------------------------------------------------
extern "C" void kernel_launch(void* const* d_in, const int* in_sizes, int n_in,
                              void* d_out, int out_size, void* d_ws,
                              size_t ws_size, hipStream_t stream) {
  const float* inputs = (const float*)d_in[0];
  const float* slots_init = (const float*)d_in[1];
  const float* ln_in_w = (const float*)d_in[2];
  const float* ln_in_b = (const float*)d_in[3];
  const float* ln_slot_w = (const float*)d_in[4];
  const float* ln_slot_b = (const float*)d_in[5];
  const float* ln_mlp_w = (const float*)d_in[6];
  const float* ln_mlp_b = (const float*)d_in[7];
  const float* Wq = (const float*)d_in[8];
  const float* Wk = (const float*)d_in[9];
  const float* Wv = (const float*)d_in[10];
  const float* gWih = (const float*)d_in[11];
  const float* gWhh = (const float*)d_in[12];
  const float* gbih = (const float*)d_in[13];
  const float* gbhh = (const float*)d_in[14];
  const float* W1 = (const float*)d_in[15];
  const float* b1 = (const float*)d_in[16];
  const float* W2 = (const float*)d_in[17];
  const float* b2 = (const float*)d_in[18];
  const float* smu = (const float*)d_in[19];
  const float* sls = (const float*)d_in[20];
  float* out = (float*)d_out;

  float* kbuf = (float*)d_ws;                         // 32*4096*128 f32
  float* vbuf = kbuf + (size_t)32 * 4096 * 128;       // 32*4096*128 f32
  float* attn = vbuf + (size_t)32 * 4096 * 128;       // 32*4096*8 f32
  float* slots = attn + (size_t)32 * 4096 * 8;        // 32768 f32
  float* qb = slots + 32768;                          // 32768 f32
  float* ub = qb + 32768;                             // 32768 f32

  k_ln_kv<<<2048, 256, 0, stream>>>(inputs, ln_in_w, ln_in_b, Wk, Wv, kbuf,
                                    vbuf);
  k_init_slots<<<128, 256, 0, stream>>>(slots_init, smu, sls, slots);
  for (int it = 0; it < 3; ++it) {
    k_q<<<16, 256, 0, stream>>>(slots, ln_slot_w, ln_slot_b, Wq, qb);
    k_attn<<<512, 256, 0, stream>>>(kbuf, qb, attn);
    k_updates<<<32, 256, 0, stream>>>(attn, vbuf, ub);
    float* dst = (it == 2) ? out : slots;
    k_gru_mlp<<<256, 128, 0, stream>>>(ub, slots, gWih, gWhh, gbih, gbhh,
                                       ln_mlp_w, ln_mlp_b, W1, b1, W2, b2, dst);
  }
}